// KnotEntangle_27762668601867
// MI455X (gfx1250) — compile-verified
//
#include <hip/hip_runtime.h>

#define Bn 8
#define Kn 32
#define Sn 256
#define Ln 8
#define Rn (Bn*Kn)   // 256 rows (b,k)

typedef __attribute__((ext_vector_type(2))) float v2f;
typedef __attribute__((ext_vector_type(8))) float v8f;

// ---- workspace layout (float offsets) -------------------------------------
enum : int {
  OFF_DFTC  = 0,                     // [S,S] cos(2*pi*s*m/S)
  OFF_DFTS  = OFF_DFTC + Sn*Sn,      // [S,S] sin(2*pi*s*m/S)
  OFF_SMEAR = OFF_DFTS + Sn*Sn,      // [R,S]
  OFF_SIGRE = OFF_SMEAR + Rn*Sn,     // [R,S]
  OFF_SIGIM = OFF_SIGRE + Rn*Sn,     // [R,S]
  OFF_ROWSUM= OFF_SIGIM + Rn*Sn,     // [R]
  OFF_MIX   = OFF_ROWSUM + Rn,       // [B,K,K]
  OFF_ROWMIX= OFF_MIX + Bn*Kn*Kn,    // [R]  sum_{j!=i}(1-mix)
  OFF_ATRE  = OFF_ROWMIX + Rn,       // [R,S]  Atilde
  OFF_ATIM  = OFF_ATRE + Rn*Sn,
  OFF_URE   = OFF_ATIM + Rn*Sn,      // [R,S]  u = sig @ P
  OFF_UIM   = OFF_URE + Rn*Sn,
  OFF_VRE   = OFF_UIM + Rn*Sn,       // [R,S]  v = Atilde @ P^T
  OFF_VIM   = OFF_VRE + Rn*Sn,
  OFF_FRE   = OFF_VIM + Rn*Sn,       // [B,S]
  OFF_FIM   = OFF_FRE + Bn*Sn,
  OFF_G     = OFF_FIM + Bn*Sn,       // [B,S]
  OFF_END   = OFF_G + Bn*Sn          // ~736K floats = ~2.9 MB
};

// f32 WMMA 16x16x4 wrapper: D = A(16x4) * B(4x16) + C
__device__ __forceinline__ v8f wmma4(v2f a, v2f b, v8f c) {
  return __builtin_amdgcn_wmma_f32_16x16x4_f32(
      /*neg_a=*/false, a, /*neg_b=*/false, b,
      /*c_mod=*/(short)0, c, /*reuse_a=*/false, /*reuse_b=*/false);
}

// Async global -> LDS copy of 16 bytes (per lane), ASYNCcnt-tracked.
// GVS mode: 64-bit SGPR base + 32-bit per-lane byte offset.
__device__ __forceinline__ void async_b128(const void* saddr, unsigned goff_bytes,
                                           const void* lds_dst) {
  unsigned loff = (unsigned)(uintptr_t)lds_dst;   // low 32 bits = LDS offset
  asm volatile("global_load_async_to_lds_b128 %0, %1, %2"
               :: "v"(loff), "v"(goff_bytes), "s"(saddr)
               : "memory");
}
__device__ __forceinline__ void wait_async0() {
  asm volatile("s_wait_asynccnt 0x0" ::: "memory");
}

// ---- kernel 0: DFT twiddle tables -----------------------------------------
__global__ void k_tables(float* ws) {
  int idx = blockIdx.x * blockDim.x + threadIdx.x;   // 0..65535
  int s = idx >> 8, m = idx & 255;
  float ang = 6.28318530717958647692f * (float)((s * m) & (Sn - 1)) / (float)Sn;
  ws[OFF_DFTC + idx] = __cosf(ang);
  ws[OFF_DFTS + idx] = __sinf(ang);
}

// ---- kernel 1: smear[b,k,s] + row sums ------------------------------------
__global__ void k_smear(const float* __restrict__ x, const float* __restrict__ sw,
                        const float* __restrict__ kmean, const float* __restrict__ klow,
                        const float* __restrict__ khigh, float* ws) {
  __shared__ float red[Sn];
  int bk = blockIdx.x;            // b*K + k
  int s  = threadIdx.x;
  int k  = bk & (Kn - 1);
  float lower = sw[0], upper = sw[1];
  float xv = x[bk];
  float xStep = (upper - lower) * xv / (float)Sn;
  float xLow  = (1.0f - lower) * xv;
  float t = xStep * ((float)(s + 1) / (float)Sn) + xLow;
  float acc = 0.0f;
#pragma unroll
  for (int l = 0; l < Ln; ++l) {
    float mean = kmean[k * Ln + l];
    float lsd  = (t <= mean) ? klow[k * Ln + l] : khigh[k * Ln + l];
    float d = (t - mean) / __expf(lsd);
    acc += __expf(-0.5f * d * d);
  }
  ws[OFF_SMEAR + bk * Sn + s] = acc;
  red[s] = acc;
  __syncthreads();
  for (int o = Sn / 2; o > 0; o >>= 1) {
    if (s < o) red[s] += red[s + o];
    __syncthreads();
  }
  if (s == 0) ws[OFF_ROWSUM + bk] = red[0];   // == sig[b,k,0].real
}

// ---- kernel 2: forward DFT as fp32 WMMA GEMM ------------------------------
// sigRe = smear @ C ; sigIm = -(smear @ Sin).  M=N=K=256.  One wave / 16x16 tile.
__global__ void k_dft(float* ws) {
  const float* A  = ws + OFF_SMEAR;
  const float* Ct = ws + OFF_DFTC;
  const float* St = ws + OFF_DFTS;
  int wave = threadIdx.x >> 5;
  int lane = threadIdx.x & 31;
  int tile = blockIdx.x * 4 + wave;            // 0..255
  int row0 = (tile >> 4) << 4;
  int col0 = (tile & 15) << 4;
  int hi = lane >> 4, ln = lane & 15;
  v8f cRe = {0.f,0.f,0.f,0.f,0.f,0.f,0.f,0.f};
  v8f cIm = {0.f,0.f,0.f,0.f,0.f,0.f,0.f,0.f};
  for (int kk = 0; kk < Sn; kk += 4) {
    int ka = kk + 2 * hi;                       // lane's K base (even)
    v2f a = *(const v2f*)(A + (row0 + ln) * Sn + ka);
    v2f bc, bs;
    bc[0] = Ct[(ka + 0) * Sn + col0 + ln];
    bc[1] = Ct[(ka + 1) * Sn + col0 + ln];
    bs[0] = St[(ka + 0) * Sn + col0 + ln];
    bs[1] = St[(ka + 1) * Sn + col0 + ln];
    cRe = wmma4(a, bc, cRe);
    cIm = wmma4(a, bs, cIm);
  }
  float* oRe = ws + OFF_SIGRE;
  float* oIm = ws + OFF_SIGIM;
#pragma unroll
  for (int r = 0; r < 8; ++r) {
    int row = row0 + r + 8 * hi;
    oRe[row * Sn + col0 + ln] =  cRe[r];
    oIm[row * Sn + col0 + ln] = -cIm[r];
  }
}

// ---- kernel 3: mix[b,i,j] from row sums -----------------------------------
__global__ void k_mix(const float* __restrict__ em, const float* __restrict__ el,
                      const float* __restrict__ eh, float* ws) {
  int idx = blockIdx.x * blockDim.x + threadIdx.x;   // 0..8191
  int b = idx >> 10, rem = idx & 1023;
  int i = rem >> 5, j = rem & 31;
  float corr = ws[OFF_ROWSUM + b * Kn + i] * ws[OFF_ROWSUM + b * Kn + j] * (1.0f / (float)Sn);
  float mean = em[j];
  float lsd  = (corr <= mean) ? el[j] : eh[j];
  float d = (corr - mean) / __expf(lsd);
  ws[OFF_MIX + idx] = __expf(-0.5f * d * d);
  (void)i;
}

// ---- kernel 4: rowmix[b,i] = sum_{j!=i}(1-mix[b,i,j]) ---------------------
__global__ void k_rowmix(float* ws) {
  int bk = threadIdx.x;               // 0..255
  int b = bk >> 5, i = bk & 31;
  const float* mix = ws + OFF_MIX + b * Kn * Kn + i * Kn;
  float acc = 0.0f;
  for (int j = 0; j < Kn; ++j)
    if (j != i) acc += 1.0f - mix[j];
  ws[OFF_ROWMIX + bk] = acc;
}

// ---- kernel 5: Atilde[b,j,m] = sum_{i!=j} mix[b,i,j]*sig[b,i,m] -----------
__global__ void k_atilde(float* ws) {
  int bj = blockIdx.x;                // b*K + j
  int b = bj >> 5, j = bj & 31;
  int m = threadIdx.x;
  const float* mix = ws + OFF_MIX + b * Kn * Kn;
  const float* sR  = ws + OFF_SIGRE;
  const float* sI  = ws + OFF_SIGIM;
  float re = 0.f, im = 0.f;
#pragma unroll 4
  for (int i = 0; i < Kn; ++i) {
    float w = mix[i * Kn + j];
    re += w * sR[(b * Kn + i) * Sn + m];
    im += w * sI[(b * Kn + i) * Sn + m];
  }
  float wd = mix[j * Kn + j];
  re -= wd * sR[bj * Sn + m];
  im -= wd * sI[bj * Sn + m];
  ws[OFF_ATRE + bj * Sn + m] = re;
  ws[OFF_ATIM + bj * Sn + m] = im;
}

// ---- kernel 6: complex WMMA GEMMs  u = sig@P[j],  v = Atilde@P[j]^T -------
// One block per j (512 threads = 16 waves; wave w owns N-tile n0 = 16*w).
// P[j] slabs are staged global->LDS with GLOBAL_LOAD_ASYNC_TO_LDS_B128
// (double-buffered, 8 K-values per slab); WMMA consumes ds_load fragments.
#define KS 8                         // K-values per slab
__global__ void __launch_bounds__(512)
k_up(const float* __restrict__ Pre_g, const float* __restrict__ Pim_g, float* ws) {
  __shared__ float sA[4 * Bn * Sn];          // [arr][b][k] : sigRe,sigIm,AtRe,AtIm
  __shared__ float uP[2][2][KS * Sn];        // [buf][c][kloc*256 + n]
  __shared__ float vP[2][2][Sn * KS];        // [buf][c][m*KS + kloc]

  int tid  = threadIdx.x;
  int wave = tid >> 5;
  int lane = tid & 31;
  int j    = blockIdx.x;
  int n0   = wave << 4;
  int hi = lane >> 4, ln = lane & 15;
  int bb = ln & 7;                            // pad M=8 -> 16 (dup rows)
  const float* PreJ = Pre_g + j * Sn * Sn;
  const float* PimJ = Pim_g + j * Sn * Sn;

  // ---- stage A-side rows (sig / Atilde for this j) into LDS: 4 x B128/thread
  {
    int b  = tid >> 6;                        // 0..7
    int kq = (tid & 63) << 2;                 // 0,4,..,252
    int rowg = (b * Kn + j) * Sn + kq;        // element offset in [R,S] arrays
    async_b128(ws, (unsigned)(OFF_SIGRE + rowg) * 4u, &sA[0 * 2048 + b * Sn + kq]);
    async_b128(ws, (unsigned)(OFF_SIGIM + rowg) * 4u, &sA[1 * 2048 + b * Sn + kq]);
    async_b128(ws, (unsigned)(OFF_ATRE  + rowg) * 4u, &sA[2 * 2048 + b * Sn + kq]);
    async_b128(ws, (unsigned)(OFF_ATIM  + rowg) * 4u, &sA[3 * 2048 + b * Sn + kq]);
  }

  // ---- slab issue: 4 x B128/thread covers uP(re,im) + vP(re,im) for KS rows
  auto issue_slab = [&](int buf, int kk8) {
    // q=0/1: uP  (rows kk8..kk8+7 of P, all 256 cols; contiguous)
    int kloc = tid >> 6;                      // 0..7
    int nq   = (tid & 63) << 2;
    unsigned gu = (unsigned)((kk8 + kloc) * Sn + nq) * 4u;
    async_b128(PreJ, gu, &uP[buf][0][kloc * Sn + nq]);
    async_b128(PimJ, gu, &uP[buf][1][kloc * Sn + nq]);
    // q=2/3: vP  (all 256 rows of P, cols kk8..kk8+7; 2 x B128 per row)
    int m    = tid >> 1;                      // 0..255
    int half = (tid & 1) << 2;                // 0 or 4
    unsigned gv = (unsigned)(m * Sn + kk8 + half) * 4u;
    async_b128(PreJ, gv, &vP[buf][0][m * KS + half]);
    async_b128(PimJ, gv, &vP[buf][1][m * KS + half]);
  };

  issue_slab(0, 0);

  v8f uRe = {0.f,0.f,0.f,0.f,0.f,0.f,0.f,0.f};
  v8f uIm = uRe, vRe = uRe, vIm = uRe;
  int cur = 0;
  for (int step = 0; step < Sn / KS; ++step) {
    wait_async0();
    __syncthreads();                          // slab `cur` (and sA) ready
    if (step + 1 < Sn / KS) issue_slab(cur ^ 1, (step + 1) * KS);
    int kk8 = step * KS;
#pragma unroll
    for (int t = 0; t < 2; ++t) {
      int kl   = t * 4 + 2 * hi;              // local K base (even)
      int kabs = kk8 + kl;
      v2f aSr = *(const v2f*)(&sA[0 * 2048 + bb * Sn + kabs]);
      v2f aSi = *(const v2f*)(&sA[1 * 2048 + bb * Sn + kabs]);
      v2f aAr = *(const v2f*)(&sA[2 * 2048 + bb * Sn + kabs]);
      v2f aAi = *(const v2f*)(&sA[3 * 2048 + bb * Sn + kabs]);
      v2f aSiN = {-aSi[0], -aSi[1]};
      v2f aAiN = {-aAi[0], -aAi[1]};
      v2f bPr, bPi;
      bPr[0] = uP[cur][0][(kl + 0) * Sn + n0 + ln];
      bPr[1] = uP[cur][0][(kl + 1) * Sn + n0 + ln];
      bPi[0] = uP[cur][1][(kl + 0) * Sn + n0 + ln];
      bPi[1] = uP[cur][1][(kl + 1) * Sn + n0 + ln];
      v2f bTr = *(const v2f*)(&vP[cur][0][(n0 + ln) * KS + kl]);
      v2f bTi = *(const v2f*)(&vP[cur][1][(n0 + ln) * KS + kl]);
      // u = sig * P (complex)
      uRe = wmma4(aSr,  bPr, uRe);
      uRe = wmma4(aSiN, bPi, uRe);
      uIm = wmma4(aSr,  bPi, uIm);
      uIm = wmma4(aSi,  bPr, uIm);
      // v = Atilde * P^T (complex)
      vRe = wmma4(aAr,  bTr, vRe);
      vRe = wmma4(aAiN, bTi, vRe);
      vIm = wmma4(aAr,  bTi, vIm);
      vIm = wmma4(aAi,  bTr, vIm);
    }
    __syncthreads();                          // done reading slab `cur`
    cur ^= 1;
  }

  if (hi == 0) {                              // rows M=0..7 are valid batches
#pragma unroll
    for (int r = 0; r < 8; ++r) {
      int o = (r * Kn + j) * Sn + n0 + ln;    // b = r
      ws[OFF_URE + o] = uRe[r];
      ws[OFF_UIM + o] = uIm[r];
      ws[OFF_VRE + o] = vRe[r];
      ws[OFF_VIM + o] = vIm[r];
    }
  }
}

// ---- kernel 7: F[b,m] (freq-domain pair sum) and G[b,s] -------------------
__global__ void k_F(const float* __restrict__ pol, float* ws) {
  int idx = blockIdx.x * blockDim.x + threadIdx.x;  // 0..2047
  int b = idx >> 8, m = idx & 255;
  float Fre = 0.f, Fim = 0.f;
  for (int j = 0; j < Kn; ++j) {
    float cp = __cosf(pol[j]), sp = __sinf(pol[j]);
    int o = (b * Kn + j) * Sn + m;
    float ur = ws[OFF_URE + o], ui = ws[OFF_UIM + o];
    float ar = ws[OFF_ATRE + o], ai = ws[OFF_ATIM + o];
    float sr = ws[OFF_SIGRE + o], si = ws[OFF_SIGIM + o];
    float vr = ws[OFF_VRE + o], vi = ws[OFF_VIM + o];
    Fre += cp * (ur * ar - ui * ai) + sp * (sr * vr - si * vi);
    Fim += cp * (ur * ai + ui * ar) + sp * (sr * vi + si * vr);
  }
  float G = 0.f;
  for (int i = 0; i < Kn; ++i)
    G += ws[OFF_SMEAR + (b * Kn + i) * Sn + m] * ws[OFF_ROWMIX + b * Kn + i];
  ws[OFF_FRE + idx] = Fre;
  ws[OFF_FIM + idx] = Fim;
  ws[OFF_G + idx]   = G;
}

// ---- kernel 8: iDFT (8 rows), attention gate, final output ----------------
__global__ void k_out(const float* __restrict__ x, const float* __restrict__ sw,
                      const float* __restrict__ aw, const float* __restrict__ ab,
                      const float* __restrict__ ascope, float* __restrict__ out,
                      const float* ws) {
  int idx = blockIdx.x * blockDim.x + threadIdx.x;  // 0..2047
  int b = idx >> 8, s = idx & 255;
  float rr = 0.f, ri = 0.f;
  for (int m = 0; m < Sn; ++m) {
    float fr = ws[OFF_FRE + b * Sn + m], fi = ws[OFF_FIM + b * Sn + m];
    float c = ws[OFF_DFTC + m * Sn + s], sn = ws[OFF_DFTS + m * Sn + s];
    rr += fr * c - fi * sn;
    ri += fr * sn + fi * c;
  }
  rr = rr * (1.0f / (float)Sn) + ws[OFF_G + idx];
  ri = ri * (1.0f / (float)Sn);
  // attention gate
  float lower = sw[0], upper = sw[1];
  float mm = 0.f;
  for (int k = 0; k < Kn; ++k) mm += x[b * Kn + k];
  mm *= (1.0f / (float)Kn);
  float xi = (float)(s + 1) / (float)Sn;
  float g = 0.f;
  for (int k = 0; k < Kn; ++k) {
    float aM = x[b * Kn + k] * aw[k] + ab[k];
    float aL = (1.0f - lower * ascope[k]) * mm;
    float aH = (1.0f + upper * ascope[k]) * mm;
    float gs = (aH - aL) * xi + aL;
    float d = (gs - aM) / __expf(gs <= aM ? aL : aH);
    g += __expf(-0.5f * d * d);
  }
  out[2 * idx + 0] = g * rr;   // complex64 interleaved
  out[2 * idx + 1] = g * ri;
}

extern "C" void kernel_launch(void* const* d_in, const int* in_sizes, int n_in,
                              void* d_out, int out_size, void* d_ws, size_t ws_size,
                              hipStream_t stream) {
  (void)in_sizes; (void)n_in; (void)out_size; (void)ws_size;
  const float* x      = (const float*)d_in[0];
  const float* sw     = (const float*)d_in[1];
  const float* kmean  = (const float*)d_in[2];
  const float* klow   = (const float*)d_in[3];
  const float* khigh  = (const float*)d_in[4];
  const float* em     = (const float*)d_in[5];
  const float* el     = (const float*)d_in[6];
  const float* eh     = (const float*)d_in[7];
  const float* pol    = (const float*)d_in[8];
  const float* pre    = (const float*)d_in[9];
  const float* pim    = (const float*)d_in[10];
  const float* aw     = (const float*)d_in[11];
  const float* ab     = (const float*)d_in[12];
  const float* ascope = (const float*)d_in[13];
  float* ws  = (float*)d_ws;
  float* out = (float*)d_out;

  k_tables<<<Sn * Sn / 256, 256, 0, stream>>>(ws);
  k_smear<<<Rn, Sn, 0, stream>>>(x, sw, kmean, klow, khigh, ws);
  k_dft<<<64, 128, 0, stream>>>(ws);
  k_mix<<<Bn * Kn * Kn / 256, 256, 0, stream>>>(em, el, eh, ws);
  k_rowmix<<<1, Rn, 0, stream>>>(ws);
  k_atilde<<<Rn, Sn, 0, stream>>>(ws);
  k_up<<<Kn, 512, 0, stream>>>(pre, pim, ws);
  k_F<<<Bn * Sn / 256, 256, 0, stream>>>(pol, ws);
  k_out<<<Bn * Sn / 256, 256, 0, stream>>>(x, sw, aw, ab, ascope, out, ws);
}